// LengthRegulator_25305947308855
// MI455X (gfx1250) — compile-verified
//
#include <hip/hip_runtime.h>
#include <stdint.h>

// Problem constants (match reference: B=16, S=512, H=512, MAX_LEN=3072)
#define B_       16
#define S_       512
#define H_       512
#define MAXLEN_  3072
#define ROWS_    16           // output frames per block
#define CPT_     128          // threads per copy block: 128 lanes * 16B = 2KB = one H row

typedef float v4f __attribute__((ext_vector_type(4)));
typedef int   v4i __attribute__((ext_vector_type(4)));

// ---------------- CDNA5 async global<->LDS path (guarded) ----------------
#if __has_builtin(__builtin_amdgcn_global_load_async_to_lds_b128) && \
    __has_builtin(__builtin_amdgcn_global_store_async_from_lds_b128) && \
    __has_builtin(__builtin_amdgcn_s_wait_asynccnt)
#define USE_ASYNC_LDS 1
#else
#define USE_ASYNC_LDS 0
#endif

#if USE_ASYNC_LDS
typedef __attribute__((address_space(1))) v4i GLv4i;   // global (AS1) v4i
typedef __attribute__((address_space(3))) v4i LDSv4i;  // LDS (AS3) v4i

__device__ __forceinline__ GLv4i* to_global(const void* p) {
    return reinterpret_cast<GLv4i*>(reinterpret_cast<uintptr_t>(p));
}
__device__ __forceinline__ LDSv4i* to_lds(const void* p) {
    // Low 32 bits of a generic pointer to LDS are the LDS byte offset (ISA 10.2).
    return reinterpret_cast<LDSv4i*>((uint32_t)reinterpret_cast<uintptr_t>(p));
}
__device__ __forceinline__ void async_g2l_b128(const void* g, void* l) {
    __builtin_amdgcn_global_load_async_to_lds_b128(to_global(g), to_lds(l), 0, 0);
}
__device__ __forceinline__ void async_l2g_b128(void* g, const void* l) {
    __builtin_amdgcn_global_store_async_from_lds_b128(to_global(g), to_lds(l), 0, 0);
}
__device__ __forceinline__ void wait_async0() {
    __builtin_amdgcn_s_wait_asynccnt(0);
}
#endif

// ---------------- Kernel: searchsorted + row replication (async data mover) --
// One block handles ROWS_ consecutive output frames of one batch. Each lane
// owns a private 16B slice of the 2KB row and moves it global -> LDS -> global
// entirely through the CDNA5 async engine (ASYNCcnt), double-buffered.
__global__ void __launch_bounds__(CPT_)
lr_regulate_kernel(const float* __restrict__ x, const int* __restrict__ cum,
                   float* __restrict__ out) {
    __shared__ int scum[S_];            // 2 KB: cum row for this batch
    __shared__ int ssrc[ROWS_];         // source index per output row (-1 = invalid)
    __shared__ v4f sbuf[2][H_ / 4];     // 2 x 2 KB double buffer for async bounce

    const int tid = threadIdx.x;        // 0..127, owns bytes [tid*16, tid*16+16)
    const int b   = blockIdx.y;
    const int t0  = blockIdx.x * ROWS_;

    // Stage cum[b][0..511] into LDS: 128 lanes x 4 ints (16B) each.
#if USE_ASYNC_LDS
    async_g2l_b128(cum + (size_t)b * S_ + tid * 4, &scum[tid * 4]);
    wait_async0();
#else
    ((v4i*)scum)[tid] = ((const v4i*)(cum + (size_t)b * S_))[tid];
#endif
    __syncthreads();

    const int total = scum[S_ - 1];
    if (tid < ROWS_) {
        const int t = t0 + tid;
        // upper_bound: first s with cum[s] > t  (searchsorted side='right')
        int lo = 0, hi = S_;
        while (lo < hi) {
            int mid = (lo + hi) >> 1;
            if (scum[mid] <= t) lo = mid + 1; else hi = mid;
        }
        if (lo > S_ - 1) lo = S_ - 1;   // clip
        ssrc[tid] = (t < total) ? lo : -1;
    }
    __syncthreads();

#if USE_ASYNC_LDS
    // Row validity is block-uniform (no divergence); lane slices are private,
    // so only per-wave s_wait_asynccnt is needed -- no barriers in the loop.
    if (ssrc[0] >= 0)
        async_g2l_b128(x + ((size_t)b * S_ + ssrc[0]) * H_ + tid * 4, &sbuf[0][tid]);
    #pragma unroll
    for (int r = 0; r < ROWS_; ++r) {
        const int src = ssrc[r];
        float* dst = out + ((size_t)b * MAXLEN_ + (t0 + r)) * H_ + tid * 4;
        wait_async0();                          // load r landed; store r-1 retired
        if (src >= 0) {
            async_l2g_b128(dst, &sbuf[r & 1][tid]);
        } else {
            v4f z = {0.f, 0.f, 0.f, 0.f};
            __builtin_nontemporal_store(z, (v4f*)dst);
        }
        if (r + 1 < ROWS_) {
            const int s2 = ssrc[r + 1];
            if (s2 >= 0)
                async_g2l_b128(x + ((size_t)b * S_ + s2) * H_ + tid * 4,
                               &sbuf[(r + 1) & 1][tid]);
        }
    }
    // s_endpgm performs an implicit wait-idle; outstanding async stores drain.
#else
    // Register fallback: B128 load + non-temporal B128 store.
    #pragma unroll
    for (int r = 0; r < ROWS_; ++r) {
        const int src = ssrc[r];
        float* dst = out + ((size_t)b * MAXLEN_ + (t0 + r)) * H_ + tid * 4;
        v4f v = {0.f, 0.f, 0.f, 0.f};
        if (src >= 0)
            v = *(const v4f*)(x + ((size_t)b * S_ + src) * H_ + tid * 4);
        __builtin_nontemporal_store(v, (v4f*)dst);
    }
#endif
}

// ---------------- Kernel: per-batch inclusive scan of rounded durations -----
__global__ void __launch_bounds__(S_)
lr_cum_kernel(const float* __restrict__ dur, int* __restrict__ cum) {
    __shared__ int sc[S_];
    const int b = blockIdx.x;
    const int i = threadIdx.x;
    sc[i] = (int)rintf(dur[b * S_ + i]);   // rintf == round-half-even == jnp.round
    __syncthreads();
    #pragma unroll
    for (int off = 1; off < S_; off <<= 1) {
        int v = (i >= off) ? sc[i - off] : 0;
        __syncthreads();
        sc[i] += v;
        __syncthreads();
    }
    cum[b * S_ + i] = sc[i];
}

// ---------------- Launch --------------------------------------------------
extern "C" void kernel_launch(void* const* d_in, const int* in_sizes, int n_in,
                              void* d_out, int out_size, void* d_ws, size_t ws_size,
                              hipStream_t stream) {
    const float* x   = (const float*)d_in[0];   // [B, S, H] f32
    const float* dur = (const float*)d_in[1];   // [B, S]    f32
    (void)in_sizes; (void)n_in; (void)out_size; (void)ws_size;

    int* cum = (int*)d_ws;                      // B*S ints = 32 KB scratch

    lr_cum_kernel<<<B_, S_, 0, stream>>>(dur, cum);

    dim3 grid(MAXLEN_ / ROWS_, B_);
    lr_regulate_kernel<<<grid, CPT_, 0, stream>>>(x, cum, (float*)d_out);
}